// DirNIWNet_88656714924609
// MI455X (gfx1250) — compile-verified
//
#include <hip/hip_runtime.h>
#include <math.h>

typedef __attribute__((ext_vector_type(2))) float v2f;
typedef __attribute__((ext_vector_type(8))) float v8f;

#define BB 32
#define NN 2048
#define DD 256
#define PP 16
#define TAUF 1.0f
#define NUM_ITERS 3
#define LOG2PI 1.8378770664093453f

__device__ __forceinline__ float softplus_f(float x) {
    return (x > 20.0f) ? x : log1pf(expf(x));
}

// ds_swizzle_b32 xor patterns (group-of-32 mode: offset = (xor_mask<<10)|0x1f)
#define SWZ_XOR1  0x041f
#define SWZ_XOR2  0x081f
#define SWZ_XOR4  0x101f
#define SWZ_XOR8  0x201f
#define SWZ_XOR16 0x401f
#define SWZF(x, imm) __int_as_float(__builtin_amdgcn_ds_swizzle(__float_as_int(x), (imm)))

// ---------------------------------------------------------------------------
// init: out <- { pi = 1/P, mu = m, Sigma = softplus(V_) } broadcast over B
// ---------------------------------------------------------------------------
__global__ void init_kernel(const float* __restrict__ m, const float* __restrict__ Vr,
                            float* __restrict__ out_pi, float* __restrict__ out_mu,
                            float* __restrict__ out_Sig) {
    int idx = blockIdx.x * blockDim.x + threadIdx.x;   // over B*P*D
    if (idx >= BB * PP * DD) return;
    int d = idx % DD;
    int p = (idx / DD) % PP;
    float mv = m[p * DD + d];
    float V  = softplus_f(Vr[p * DD + d]);
    out_mu[idx]  = mv;
    out_Sig[idx] = V;
    if (d == 0) out_pi[idx / DD] = 1.0f / PP;          // idx/DD == b*P + p
}

// ---------------------------------------------------------------------------
// params: from (pi, mu, Sigma) build E-step operands:
//   W2[b][d][p] = ( -0.5/Sigma , mu/Sigma )       (float2, K-pair per dim)
//   Cc[b][p]    = log(pi) - 0.5*(D*log2pi + sum_d log Sigma + sum_d mu^2/Sigma)
// ---------------------------------------------------------------------------
__global__ __launch_bounds__(DD) void params_kernel(
        const float* __restrict__ out_pi, const float* __restrict__ out_mu,
        const float* __restrict__ out_Sig,
        float2* __restrict__ W2, float* __restrict__ Cc) {
    __shared__ float red[DD];
    int blk = blockIdx.x;
    int b = blk / PP, p = blk % PP;
    int d = threadIdx.x;
    float mu   = out_mu [(size_t)(b * PP + p) * DD + d];
    float S    = out_Sig[(size_t)(b * PP + p) * DD + d];
    float invS = 1.0f / S;
    W2[((size_t)b * DD + d) * PP + p] = make_float2(-0.5f * invS, mu * invS);
    red[d] = logf(S) + mu * mu * invS;
    __syncthreads();
    for (int s = DD / 2; s > 0; s >>= 1) {
        if (d < s) red[d] += red[d + s];
        __syncthreads();
    }
    if (d == 0)
        Cc[b * PP + p] = logf(out_pi[b * PP + p]) - 0.5f * (DD * LOG2PI + red[0]);
}

// ---------------------------------------------------------------------------
// E-step: per 16-row tile of data, jll[16n x 16p] via f32 WMMA over K=2D=512,
// per-row logsumexp over P via ds_swizzle xor-reduction, qq stored transposed.
// Weight matrix staged global->LDS with CDNA5 async-to-LDS b128 copies.
// blockDim=128 (4 waves, one n-tile each); 32 blocks per batch.
// ---------------------------------------------------------------------------
__global__ __launch_bounds__(128) void estep_kernel(
        const float* __restrict__ data, const float* __restrict__ mask,
        const float2* __restrict__ W2, const float* __restrict__ Cc,
        float* __restrict__ qqT) {
    __shared__ __align__(16) float2 sW[DD * PP];   // 32 KB: weights for batch b
    int tid    = threadIdx.x;
    int b      = blockIdx.x >> 5;                  // 32 blocks per batch
    int blkInB = blockIdx.x & 31;

    // ---- async global->LDS staging of W2[b] (2048 x b128, 16 per thread) ----
    {
        unsigned long long gbase =
            (unsigned long long)(uintptr_t)(W2 + (size_t)b * DD * PP);
        unsigned int ldsbase = (unsigned int)(uintptr_t)&sW[0];  // low 32 = LDS off
        #pragma unroll
        for (int i = 0; i < 16; ++i) {
            unsigned int voff = (unsigned int)(tid + i * 128) * 16u;
            unsigned int ldsa = ldsbase + voff;
            asm volatile("global_load_async_to_lds_b128 %0, %1, %2"
                         :: "v"(ldsa), "v"(voff), "s"(gbase) : "memory");
        }
        asm volatile("s_wait_asynccnt 0x0" ::: "memory");
    }
    __syncthreads();

    int wave = tid >> 5, lane = tid & 31;
    int half = lane >> 4, lp = lane & 15;
    int n0 = (blkInB * 4 + wave) * 16;

    // A operand: lane reads x = data[b][n0+lp][2s+half], a = (x*x, x)
    const float* Arow = data + ((size_t)b * NN + n0 + lp) * DD + half;

    v8f c = {};
    #pragma unroll 8
    for (int s = 0; s < DD / 2; ++s) {
        float x = Arow[2 * s];
        v2f a;  a.x = x * x; a.y = x;
        float2 w = sW[(2 * s + half) * PP + lp];
        v2f bv; bv.x = w.x;  bv.y = w.y;
        c = __builtin_amdgcn_wmma_f32_16x16x4_f32(false, a, false, bv,
                                                  (short)0, c, false, false);
    }

    float cadd = Cc[b * PP + lp];             // column p = lp
    float out[8];
    #pragma unroll
    for (int r = 0; r < 8; ++r) {
        float v = c[r] + cadd;                // jll for row n0+r+8*half, col lp
        float mx = v;                         // max over the 16 p-lanes
        mx = fmaxf(mx, SWZF(mx, SWZ_XOR1));
        mx = fmaxf(mx, SWZF(mx, SWZ_XOR2));
        mx = fmaxf(mx, SWZF(mx, SWZ_XOR4));
        mx = fmaxf(mx, SWZF(mx, SWZ_XOR8));
        float ssum = expf(v - mx);            // sum of exp over the 16 p-lanes
        ssum += SWZF(ssum, SWZ_XOR1);
        ssum += SWZF(ssum, SWZ_XOR2);
        ssum += SWZF(ssum, SWZ_XOR4);
        ssum += SWZF(ssum, SWZ_XOR8);
        float mll = mx + logf(ssum);
        int row = n0 + r + 8 * half;
        out[r] = expf(v - mll) * mask[(size_t)b * NN + row];
    }
    // rows are contiguous per lane in qqT[b][p][n]
    float* dst = qqT + ((size_t)(b * PP + lp)) * NN + n0 + 8 * half;
    ((float4*)dst)[0] = make_float4(out[0], out[1], out[2], out[3]);
    ((float4*)dst)[1] = make_float4(out[4], out[5], out[6], out[7]);
}

// ---------------------------------------------------------------------------
// M-step: wx[p][d] = sum_n qq[n][p]*x[n][d], wxx with x^2, wsum = sum_n qq.
// One wave per (batch, 16-col d-tile): 16x16 output tile, K=N=2048.
// ---------------------------------------------------------------------------
__global__ __launch_bounds__(128) void mstep_kernel(
        const float* __restrict__ data, const float* __restrict__ qqT,
        float* __restrict__ wsum, float* __restrict__ wx, float* __restrict__ wxx) {
    int tid   = threadIdx.x;
    int wglob = blockIdx.x * 4 + (tid >> 5);
    int b     = wglob >> 4;
    int dtile = wglob & 15;
    int d0    = dtile * 16;
    int lane = tid & 31, half = lane >> 4, lp = lane & 15;

    // A operand: qqT[p=lp][4s + 2*half + {0,1}]  -> one float2 per step
    const float2* ap = (const float2*)(qqT + ((size_t)(b * PP + lp)) * NN) + half;
    // B operand: x at rows n = 4s + 2*half + {0,1}, col d0+lp
    const float* bp = data + ((size_t)b * NN + 2 * half) * DD + d0 + lp;

    v8f cx = {}, cxx = {};
    float asum = 0.0f;
    #pragma unroll 4
    for (int s = 0; s < NN / 4; ++s) {
        float2 a2 = ap[2 * s];
        v2f a;   a.x = a2.x;     a.y = a2.y;
        float x0 = bp[(size_t)(4 * s) * DD];
        float x1 = bp[(size_t)(4 * s + 1) * DD];
        v2f bx;  bx.x = x0;      bx.y = x1;
        v2f bxx; bxx.x = x0*x0;  bxx.y = x1*x1;
        cx  = __builtin_amdgcn_wmma_f32_16x16x4_f32(false, a, false, bx,
                                                    (short)0, cx,  false, false);
        cxx = __builtin_amdgcn_wmma_f32_16x16x4_f32(false, a, false, bxx,
                                                    (short)0, cxx, false, false);
        asum += a2.x + a2.y;                 // responsibility sum for p=lp
    }
    #pragma unroll
    for (int r = 0; r < 8; ++r) {
        int p = r + 8 * half;
        wx [((size_t)(b * PP + p)) * DD + d0 + lp] = cx[r];
        wxx[((size_t)(b * PP + p)) * DD + d0 + lp] = cxx[r];
    }
    asum += SWZF(asum, SWZ_XOR16);           // combine the two n-halves
    if (dtile == 0 && half == 0) wsum[b * PP + lp] = asum;
}

// ---------------------------------------------------------------------------
// update: MAP M-step closed form -> (pi, mu, Sigma) into d_out
// ---------------------------------------------------------------------------
__global__ __launch_bounds__(DD) void update_kernel(
        const float* __restrict__ m, const float* __restrict__ Vr,
        const float* __restrict__ wsum, const float* __restrict__ wx,
        const float* __restrict__ wxx,
        float* __restrict__ out_pi, float* __restrict__ out_mu,
        float* __restrict__ out_Sig) {
    int blk = blockIdx.x;
    int b = blk / PP, p = blk % PP;
    int d = threadIdx.x;
    float wr = wsum[b * PP + p] + TAUF;
    float mv = m[p * DD + d];
    float V  = softplus_f(Vr[p * DD + d]);
    float mu = (wx [((size_t)(b * PP + p)) * DD + d] + mv * TAUF) / wr;
    float Sg = (wxx[((size_t)(b * PP + p)) * DD + d] + (V + mv * mv) * TAUF) / wr
               - mu * mu;
    out_mu [((size_t)(b * PP + p)) * DD + d] = mu;
    out_Sig[((size_t)(b * PP + p)) * DD + d] = Sg;
    if (d == 0) {
        float denom = 0.0f;
        for (int q = 0; q < PP; ++q) denom += wsum[b * PP + q] + TAUF;
        out_pi[b * PP + p] = wr / denom;
    }
}

// ---------------------------------------------------------------------------
extern "C" void kernel_launch(void* const* d_in, const int* in_sizes, int n_in,
                              void* d_out, int out_size, void* d_ws, size_t ws_size,
                              hipStream_t stream) {
    const float* data = (const float*)d_in[0];   // [B,N,D]
    const float* mask = (const float*)d_in[1];   // [B,N]
    const float* m    = (const float*)d_in[2];   // [P,D]
    const float* Vr   = (const float*)d_in[3];   // [P,D]
    (void)in_sizes; (void)n_in; (void)out_size;

    float* out     = (float*)d_out;
    float* out_pi  = out;                              // [B,P]
    float* out_mu  = out + BB * PP;                    // [B,P,D]
    float* out_Sig = out + BB * PP + (size_t)BB * PP * DD;

    float*  ws   = (float*)d_ws;
    float2* W2   = (float2*)ws;                        // B*D*P float2   (1 MB)
    float*  Cc   = ws + (size_t)BB * DD * PP * 2;      // B*P
    float*  qqT  = Cc + BB * PP;                       // B*P*N          (4 MB)
    float*  wsum = qqT + (size_t)BB * PP * NN;         // B*P
    float*  wx   = wsum + BB * PP;                     // B*P*D
    float*  wxx  = wx + (size_t)BB * PP * DD;          // B*P*D
    (void)ws_size;

    init_kernel<<<(BB * PP * DD + 255) / 256, 256, 0, stream>>>(
        m, Vr, out_pi, out_mu, out_Sig);

    for (int it = 0; it < NUM_ITERS; ++it) {
        params_kernel<<<BB * PP, DD, 0, stream>>>(out_pi, out_mu, out_Sig, W2, Cc);
        estep_kernel<<<BB * (NN / 16) / 4, 128, 0, stream>>>(data, mask, W2, Cc, qqT);
        mstep_kernel<<<BB * 16 / 4, 128, 0, stream>>>(data, qqT, wsum, wx, wxx);
        update_kernel<<<BB * PP, DD, 0, stream>>>(m, Vr, wsum, wx, wxx,
                                                  out_pi, out_mu, out_Sig);
    }
}